// RoI_layer_mulity_61521111548339
// MI455X (gfx1250) — compile-verified
//
#include <hip/hip_runtime.h>
#include <hip/hip_bf16.h>

// Problem constants (from reference)
#define VF      128          // V*F = 8*16
#define NCH     1024         // C
#define HW      196          // 14*14
#define NBOX    768
#define OUTP    49           // 7*7
#define CT      32           // channels per RoI block
#define LDSTR   197          // padded plane stride (197 mod 64 = 5, coprime -> conflict-free)

#define ROI_ELEMS   (38535168ull)   // 768*1024*49
#define SNODE_ELEMS (131072ull)     // 128*1024
#define FAKE_ELEMS  (18874368ull)   // 128*1024*144

typedef __attribute__((ext_vector_type(2))) float v2f;
typedef __attribute__((ext_vector_type(8))) float v8f;

// ---------------------------------------------------------------------------
// Kernel 1: RoIAlignAvg. One block = (box, 32-channel tile).
// Stage 32 planes to LDS, precompute per-box bilinear tap tables, then each
// thread produces pooled outputs with 16 LDS FMAs each.
// ---------------------------------------------------------------------------
__global__ __launch_bounds__(256) void roi_align_kernel(
    const float* __restrict__ feat, const float* __restrict__ boxes,
    const int* __restrict__ fidx, float* __restrict__ out) {
  __shared__ float planes[CT * LDSTR];
  __shared__ int   toff[64][4];
  __shared__ float twgt[64][4];
  __shared__ int   six0[8], six1[8], siy0[8], siy1[8];
  __shared__ float swx[8], swy[8];

  const int n   = blockIdx.y;
  const int c0  = blockIdx.x * CT;
  const int tid = threadIdx.x;

  // --- per-box grid coordinates (16 threads: 8 for x, 8 for y) ---
  if (tid < 16) {
    const int axis = tid >> 3;            // 0 = x, 1 = y
    const int i    = tid & 7;
    const float scale = 1.0f / 16.0f;
    float lo = boxes[n * 4 + axis]     * scale;
    float hi = boxes[n * 4 + axis + 2] * scale;
    float t  = (float)i * (1.0f / 7.0f);  // linspace(0,1,8)
    float v  = lo + t * (hi - lo);
    v = fminf(fmaxf(v, 0.0f), 13.0f);     // clip to [0, W-1]
    int   p0 = (int)floorf(v);
    int   p1 = min(p0 + 1, 13);
    float w  = v - (float)p0;
    if (axis == 0) { six0[i] = p0; six1[i] = p1; swx[i] = w; }
    else           { siy0[i] = p0; siy1[i] = p1; swy[i] = w; }
  }
  __syncthreads();

  // --- 64 grid points -> 4 taps each (offset + weight) ---
  if (tid < 64) {
    const int gy = tid >> 3, gx = tid & 7;
    const int x0 = six0[gx], x1 = six1[gx];
    const int y0 = siy0[gy], y1 = siy1[gy];
    const float wx = swx[gx], wy = swy[gy];
    toff[tid][0] = y0 * 14 + x0;  twgt[tid][0] = (1.0f - wy) * (1.0f - wx);
    toff[tid][1] = y0 * 14 + x1;  twgt[tid][1] = (1.0f - wy) * wx;
    toff[tid][2] = y1 * 14 + x0;  twgt[tid][2] = wy * (1.0f - wx);
    toff[tid][3] = y1 * 14 + x1;  twgt[tid][3] = wy * wx;
  }

  // --- stage 32 channel planes into LDS (frame is L2-resident) ---
  const size_t fb = (size_t)fidx[n] * (size_t)(NCH * HW) + (size_t)c0 * HW;
  for (int idx = tid; idx < CT * HW; idx += 256) {
    const int c = idx / HW, s = idx - c * HW;
    planes[c * LDSTR + s] = feat[fb + (size_t)c * HW + s];
  }
  __syncthreads();

  // --- compute pooled outputs: 16 taps per output, pixel-contiguous stores ---
  for (int t = tid; t < CT * OUTP; t += 256) {
    const int c = t / OUTP, pix = t - c * OUTP;
    const int i = pix / 7, j = pix - i * 7;
    const float* __restrict__ pl = &planes[c * LDSTR];
    float acc = 0.0f;
#pragma unroll
    for (int dy = 0; dy < 2; ++dy) {
#pragma unroll
      for (int dx = 0; dx < 2; ++dx) {
        const int gp = (i + dy) * 8 + (j + dx);
#pragma unroll
        for (int t4 = 0; t4 < 4; ++t4)
          acc = fmaf(twgt[gp][t4], pl[toff[gp][t4]], acc);
      }
    }
    out[((size_t)n * NCH + c0 + c) * OUTP + pix] = 0.25f * acc;
  }
}

// ---------------------------------------------------------------------------
// Kernel 2: s_node = mean over H,W via V_WMMA_F32_16X16X4_F32.
// Each wave reduces 16 rows of 196 elements: 49 K=4 WMMA steps with B == 1/196
// (uniform B makes the B VGPR layout irrelevant). D[m, n] is the row-mean
// replicated across all 16 columns; lanes 0 and 16 write out their 8 rows.
// A-layout (ISA 7.12.2, 32-bit A 16x4): lanes 0-15 hold K=0,1; lanes 16-31 K=2,3.
// ---------------------------------------------------------------------------
__global__ __launch_bounds__(256) void snode_wmma_kernel(
    const float* __restrict__ feat, float* __restrict__ out) {
  const int wave    = (int)((blockIdx.x * blockDim.x + threadIdx.x) >> 5);
  const int lane    = threadIdx.x & 31;
  const size_t rowB = (size_t)wave * 16;
  const size_t row  = rowB + (size_t)(lane & 15);
  const int khalf   = (lane >> 4) * 2;
  const float* __restrict__ src = feat + row * HW + khalf;

  const v2f b = {1.0f / 196.0f, 1.0f / 196.0f};
  v8f acc = {};
#pragma unroll 7
  for (int k = 0; k < 49; ++k) {
    v2f a = *(const v2f*)(src + k * 4);   // 8B-aligned: row*196 + {0,2} + 4k
    acc = __builtin_amdgcn_wmma_f32_16x16x4_f32(
        /*neg_a=*/false, a, /*neg_b=*/false, b,
        /*c_mod=*/(short)0, acc, /*reuse_a=*/false, /*reuse_b=*/false);
  }
  // D layout: lanes 0-15 vgpr r -> M=r; lanes 16-31 vgpr r -> M=r+8 (cols equal)
  if (lane == 0) {
#pragma unroll
    for (int r = 0; r < 8; ++r) out[rowB + r] = acc[r];
  } else if (lane == 16) {
#pragma unroll
    for (int r = 0; r < 8; ++r) out[rowB + 8 + r] = acc[r];
  }
}

// ---------------------------------------------------------------------------
// Kernel 3: fake_h_s = features[..., 1:13, 1:13] strided copy.
// ---------------------------------------------------------------------------
__global__ __launch_bounds__(256) void fake_crop_kernel(
    const float* __restrict__ feat, float* __restrict__ out) {
  const size_t total  = FAKE_ELEMS;
  const size_t stride = (size_t)gridDim.x * blockDim.x;
  for (size_t idx = (size_t)blockIdx.x * blockDim.x + threadIdx.x;
       idx < total; idx += stride) {
    const size_t p   = idx / 144;
    const int    rem = (int)(idx - p * 144);
    const int    r   = rem / 12, s = rem - r * 12;
    out[idx] = feat[p * HW + (size_t)(r + 1) * 14 + (s + 1)];
  }
}

extern "C" void kernel_launch(void* const* d_in, const int* in_sizes, int n_in,
                              void* d_out, int out_size, void* d_ws, size_t ws_size,
                              hipStream_t stream) {
  (void)in_sizes; (void)n_in; (void)out_size; (void)d_ws; (void)ws_size;
  const float* features = (const float*)d_in[0];
  const float* boxes    = (const float*)d_in[1];
  const int*   frameIdx = (const int*)d_in[2];

  float* roi   = (float*)d_out;
  float* snode = roi + ROI_ELEMS;
  float* fake  = snode + SNODE_ELEMS;

  // RoIAlign: 768 boxes x 32 channel-tiles
  roi_align_kernel<<<dim3(NCH / CT, NBOX), 256, 0, stream>>>(features, boxes, frameIdx, roi);

  // s_node: 131072 rows / 16 rows-per-wave = 8192 waves = 1024 blocks x 8 waves
  snode_wmma_kernel<<<1024, 256, 0, stream>>>(features, snode);

  // interior crop copy
  fake_crop_kernel<<<8192, 256, 0, stream>>>(features, fake);
}